// SpectralConv1d_local_51857435131964
// MI455X (gfx1250) — compile-verified
//
#include <hip/hip_runtime.h>
#include <math.h>

// MI455X / gfx1250, wave32. fp32 path via V_WMMA_F32_16X16X4_F32.
// Roofline: ~37 GFLOP fp32, ~134 MB HBM traffic -> compute-bound; WMMA fp32 required.
// Staging uses GLOBAL_LOAD_ASYNC_TO_LDS (ASYNCcnt) to avoid VGPR round-trips.

typedef __attribute__((ext_vector_type(2))) float v2f;
typedef __attribute__((ext_vector_type(8))) float v8f;

#define B_  16
#define L_  1024
#define H_  8
#define E_  64
#define M_  64
#define SCALE_ (1.0f/(512.0f*512.0f))

// workspace float offsets (1 MB total)
#define WS_DCOS 0        // [64][1024]  cos(2*pi*l*m/1024)          (fwd DFT, real)
#define WS_DSIN 65536    // [64][1024] -sin(2*pi*l*m/1024)          (fwd DFT, imag)
#define WS_ICOS 131072   // [64][1024]  c_m*cos/1024                (irfft, real coeff)
#define WS_ISIN 196608   // [64][1024] -c_m*sin/1024                (irfft, imag coeff)

__device__ __forceinline__ v8f wmma4(v2f a, v2f b, v8f c) {
  // D = A(16x4) * B(4x16) + C(16x16), fp32
  return __builtin_amdgcn_wmma_f32_16x16x4_f32(false, a, false, b, (short)0, c, false, false);
}

__device__ __forceinline__ v8f zero8() { v8f z = {0.f,0.f,0.f,0.f,0.f,0.f,0.f,0.f}; return z; }

// --- CDNA5 async global->LDS copies (ASYNCcnt). LDS offset = low 32 bits of
// the flat pointer (ISA: LDS aperture address[31:0] is the LDS byte address).
__device__ __forceinline__ void async_g2l_b128(const float* smp, const float* g) {
  unsigned lds = (unsigned)(uintptr_t)smp;
  asm volatile("global_load_async_to_lds_b128 %0, %1, off"
               :: "v"(lds), "v"(g) : "memory");
}
__device__ __forceinline__ void async_g2l_b32(const float* smp, const float* g) {
  unsigned lds = (unsigned)(uintptr_t)smp;
  asm volatile("global_load_async_to_lds_b32 %0, %1, off"
               :: "v"(lds), "v"(g) : "memory");
}
__device__ __forceinline__ void wait_async0() {
  asm volatile("s_wait_asynccnt 0x0" ::: "memory");
}

// ---------------------------------------------------------------------------
// Kernel 0: build DFT / iDFT basis tables in workspace (deterministic each call)
// ---------------------------------------------------------------------------
__global__ __launch_bounds__(256) void twiddle_init(float* __restrict__ ws) {
  int idx = blockIdx.x * 256 + threadIdx.x;   // 65536 = 64 modes * 1024 l
  int l = idx & 1023;
  int m = idx >> 10;
  // exact-range argument reduction: angle depends only on (l*m) mod 1024
  float ang = (6.28318530717958647692f / 1024.0f) * (float)((l * m) & 1023);
  float c = cosf(ang);
  float s = sinf(ang);
  float cm = (m == 0) ? 1.0f : 2.0f;          // Hermitian doubling for m>=1
  ws[WS_DCOS + m * 1024 + l] = c;
  ws[WS_DSIN + m * 1024 + l] = -s;
  ws[WS_ICOS + m * 1024 + l] = cm * c * (1.0f / 1024.0f);
  ws[WS_ISIN + m * 1024 + l] = -cm * s * (1.0f / 1024.0f);
}

// ---------------------------------------------------------------------------
// Kernel 1: spectral half. One block per (b,h). 256 threads = 8 waves.
//   P1: X[e,m] (real+imag) = GEMM over L=1024 (WMMA f32, K-steps of 4)
//   P2: Y[o,m] = sum_i X[i,m]*w[h,i,o,m] (complex, per-mode; VALU)
//   P3: x_spec[o,l] = Yr@icos + Yi@isin (WMMA f32), write 0.5*x_spec to d_out
// ---------------------------------------------------------------------------
__global__ __launch_bounds__(256)
void spectral_kernel(const float* __restrict__ q,
                     const float* __restrict__ wre,
                     const float* __restrict__ wim,
                     const float* __restrict__ ws,
                     float* __restrict__ out) {
  __shared__ float sm[16384];                 // 64 KB, regions time-multiplexed
  const int b    = blockIdx.x >> 3;
  const int h    = blockIdx.x & 7;
  const int tid  = threadIdx.x;
  const int lane = tid & 31;
  const int wid  = tid >> 5;
  const int part = wid >> 2;                  // 0 = real(cos), 1 = imag(sin)
  const int rg   = wid & 3;                   // 16-row group of e / o
  const int mcol = lane & 15;
  const int kl   = (lane >> 4) << 1;          // A/B frag k-offset per half-wave

  // region offsets (floats)
  const int XS = 0;      // [64][66]  x[e][l] chunk
  const int CB = 4224;   // [64][68]  cos[m][l] chunk
  const int SB = 8576;   // [64][68]  sin[m][l] chunk
  const int XR = 8192;   // [64][64]  Re X   (after P1)
  const int XI = 12288;  // [64][64]  Im X
  const int YR = 0;      // [64][64]  Re Y   (after P2)
  const int YI = 4096;   // [64][64]  Im Y
  const int IC = 8192;   // [64][64]  icos chunk (P3, overwrites X)
  const int IS = 12288;  // [64][64]  isin chunk

  // ---------------- Phase 1: forward DFT as GEMM ----------------
  v8f acc[4];
#pragma unroll
  for (int t = 0; t < 4; ++t) acc[t] = zero8();

  for (int lc = 0; lc < 16; ++lc) {
    const int l0 = lc * 64;
    __syncthreads();
    {   // stage x[e][l] transposed: per-element async b32 copies
      int l  = tid & 63;
      int e0 = (tid >> 6) << 4;
      const float* src = q + ((((size_t)b * 1024 + l0 + l) * 8 + h) * 64 + e0);
#pragma unroll
      for (int j = 0; j < 16; ++j)
        async_g2l_b32(&sm[XS + (e0 + j) * 66 + l], src + j);
      // stage basis [m][l] rows: contiguous async b128 copies
      int m  = tid >> 2;
      int lo = (tid & 3) << 4;
      const float* cs = ws + WS_DCOS + m * 1024 + l0 + lo;
      const float* sn = ws + WS_DSIN + m * 1024 + l0 + lo;
#pragma unroll
      for (int j = 0; j < 4; ++j) {
        async_g2l_b128(&sm[CB + m * 68 + lo + 4 * j], cs + 4 * j);
        async_g2l_b128(&sm[SB + m * 68 + lo + 4 * j], sn + 4 * j);
      }
    }
    wait_async0();
    __syncthreads();
    const int bbase = part ? SB : CB;
    const int erow  = rg * 16 + mcol;
#pragma unroll
    for (int k0 = 0; k0 < 64; k0 += 4) {
      v2f a = *(const v2f*)&sm[XS + erow * 66 + k0 + kl];
#pragma unroll
      for (int mt = 0; mt < 4; ++mt) {
        v2f bf = *(const v2f*)&sm[bbase + (mt * 16 + mcol) * 68 + k0 + kl];
        acc[mt] = wmma4(a, bf, acc[mt]);
      }
    }
  }
  __syncthreads();
  {   // write X tiles (C layout: vgpr r -> row r (+8 for upper half-wave), col = lane%16)
    const int xbase = part ? XI : XR;
    const int rowh  = (lane >> 4) * 8;
#pragma unroll
    for (int mt = 0; mt < 4; ++mt)
#pragma unroll
      for (int r = 0; r < 8; ++r)
        sm[xbase + (rg * 16 + rowh + r) * 64 + mt * 16 + mcol] = acc[mt][r];
  }
  __syncthreads();

  // ---------------- Phase 2: per-mode complex channel mix ----------------
  {
    const int m  = tid & 63;
    const int og = tid >> 6;                  // 16-wide o-group
    float orr[16], oii[16];
#pragma unroll
    for (int j = 0; j < 16; ++j) { orr[j] = 0.f; oii[j] = 0.f; }
    for (int i = 0; i < 64; ++i) {
      float xr = sm[XR + i * 64 + m];
      float xi = sm[XI + i * 64 + m];
      const float* wrp = wre + ((((size_t)h * 64 + i) * 64 + og * 16) * 64 + m);
      const float* wip = wim + ((((size_t)h * 64 + i) * 64 + og * 16) * 64 + m);
#pragma unroll
      for (int j = 0; j < 16; ++j) {
        float a = wrp[j * 64], c = wip[j * 64];
        orr[j] += xr * a - xi * c;
        oii[j] += xr * c + xi * a;
      }
    }
#pragma unroll
    for (int j = 0; j < 16; ++j) {
      sm[YR + (og * 16 + j) * 64 + m] = orr[j];
      sm[YI + (og * 16 + j) * 64 + m] = oii[j];
    }
  }
  __syncthreads();   // Y ready; X region now dead (reused for iDFT basis)

  // ---------------- Phase 3: irfft as GEMM ----------------
  const int half = wid >> 2;
  for (int lc = 0; lc < 16; ++lc) {
    const int l0 = lc * 64;
    if (lc) __syncthreads();
    {   // stage icos/isin [m][l] chunk via async b128 copies
      int m  = tid >> 2;
      int lo = (tid & 3) << 4;
      const float* ic = ws + WS_ICOS + m * 1024 + l0 + lo;
      const float* is = ws + WS_ISIN + m * 1024 + l0 + lo;
#pragma unroll
      for (int j = 0; j < 4; ++j) {
        async_g2l_b128(&sm[IC + m * 64 + lo + 4 * j], ic + 4 * j);
        async_g2l_b128(&sm[IS + m * 64 + lo + 4 * j], is + 4 * j);
      }
    }
    wait_async0();
    __syncthreads();
    const int orow = rg * 16 + mcol;
#pragma unroll
    for (int ct2 = 0; ct2 < 2; ++ct2) {
      const int ct = half * 2 + ct2;          // l-column tile in chunk
      v8f c = zero8();
#pragma unroll
      for (int k0 = 0; k0 < 64; k0 += 4) {
        v2f ar = *(const v2f*)&sm[YR + orow * 64 + k0 + kl];
        v2f ai = *(const v2f*)&sm[YI + orow * 64 + k0 + kl];
        const int n = ct * 16 + mcol;
        v2f bc, bs;
        bc.x = sm[IC + (k0 + kl) * 64 + n];
        bc.y = sm[IC + (k0 + kl + 1) * 64 + n];
        bs.x = sm[IS + (k0 + kl) * 64 + n];
        bs.y = sm[IS + (k0 + kl + 1) * 64 + n];
        c = wmma4(ar, bc, c);
        c = wmma4(ai, bs, c);
      }
      const int rowh = (lane >> 4) * 8;
#pragma unroll
      for (int r = 0; r < 8; ++r) {
        int o = rg * 16 + rowh + r;
        int l = l0 + ct * 16 + mcol;
        out[(((size_t)b * 1024 + l) * 8 + h) * 64 + o] = 0.5f * c[r];
      }
    }
  }
}

// ---------------------------------------------------------------------------
// Kernel 2: attention half (single pass; tiny softmax temperature -> no max
// tracking needed). Block = 8 waves x 16 query rows = 128 rows; grid = B*H*8.
// d_out already holds x_spec/2; this adds V_local/2.
// ---------------------------------------------------------------------------
__global__ __launch_bounds__(256)
void attn_kernel(const float* __restrict__ q, const float* __restrict__ kk,
                 const float* __restrict__ vv, float* __restrict__ out) {
  __shared__ float sm[4096];                  // 16 KB: K tile, V tile, P patches
  const int KT = 0, VT = 1024, PT = 2048;
  const int bh   = blockIdx.x >> 3;
  const int rt   = blockIdx.x & 7;
  const int b    = bh >> 3;
  const int h    = bh & 7;
  const int tid  = threadIdx.x;
  const int lane = tid & 31;
  const int wid  = tid >> 5;
  const int qrow0 = rt * 128 + wid * 16;
  const int ncol  = lane & 15;
  const int kl    = (lane >> 4) << 1;

  // Q rows as 16 A-fragments (K=64 -> 16 k-steps of 4)
  v2f qa[16];
  {
    const int row = qrow0 + ncol;
    const float* qp = q + (((size_t)b * 1024 + row) * 8 + h) * 64;
#pragma unroll
    for (int kg = 0; kg < 16; ++kg)
      qa[kg] = *(const v2f*)(qp + kg * 4 + kl);
  }

  v8f oacc[4];
#pragma unroll
  for (int t = 0; t < 4; ++t) oacc[t] = zero8();
  float rsum[8];
#pragma unroll
  for (int r = 0; r < 8; ++r) rsum[r] = 0.f;

  const int pbase = PT + wid * 256;           // wave-private 16x16 P patch

  for (int st = 0; st < 64; ++st) {
    __syncthreads();
    {   // cooperative async stage of K/V 16x64 tiles (no VGPR round-trip)
      int s  = tid >> 4;
      int e0 = (tid & 15) << 2;
      size_t g = (((size_t)b * 1024 + st * 16 + s) * 8 + h) * 64 + e0;
      async_g2l_b128(&sm[KT + s * 64 + e0], kk + g);
      async_g2l_b128(&sm[VT + s * 64 + e0], vv + g);
    }
    wait_async0();
    __syncthreads();

    // S = Q * K^T  (16x16 tile, K=64)
    v8f sc = zero8();
#pragma unroll
    for (int kg = 0; kg < 16; ++kg) {
      v2f bf = *(const v2f*)&sm[KT + ncol * 64 + kg * 4 + kl];  // B[k=e][n=s]
      sc = wmma4(qa[kg], bf, sc);
    }

    // P = exp(SCALE*S); row-sum accumulate; spill P to LDS for layout change
    const int rowh = (lane >> 4) * 8;
#pragma unroll
    for (int r = 0; r < 8; ++r) {
      float p = __expf(sc[r] * SCALE_);
      rsum[r] += p;
      sm[pbase + (rowh + r) * 16 + ncol] = p;
    }

    // O += P * V  (K=16 -> 4 k-steps, 4 e-tiles)
#pragma unroll
    for (int k0 = 0; k0 < 16; k0 += 4) {
      v2f pa = *(const v2f*)&sm[pbase + ncol * 16 + k0 + kl];   // A[m=q][k=s]
#pragma unroll
      for (int et = 0; et < 4; ++et) {
        v2f bv;
        const int n = et * 16 + ncol;
        bv.x = sm[VT + (k0 + kl) * 64 + n];
        bv.y = sm[VT + (k0 + kl + 1) * 64 + n];
        oacc[et] = wmma4(pa, bv, oacc[et]);
      }
    }
  }

  // finish row sums (16 lanes per half-wave hold the 16 columns of each row)
#pragma unroll
  for (int r = 0; r < 8; ++r) {
    float s = rsum[r];
    s += __shfl_xor(s, 1);
    s += __shfl_xor(s, 2);
    s += __shfl_xor(s, 4);
    s += __shfl_xor(s, 8);
    rsum[r] = 0.5f / s;                       // includes the final /2
  }
  const int rowh = (lane >> 4) * 8;
#pragma unroll
  for (int et = 0; et < 4; ++et)
#pragma unroll
    for (int r = 0; r < 8; ++r) {
      int row = qrow0 + rowh + r;
      int e   = et * 16 + ncol;
      size_t idx = (((size_t)b * 1024 + row) * 8 + h) * 64 + e;
      out[idx] += oacc[et][r] * rsum[r];      // add V_local/2 onto x_spec/2
    }
}

// ---------------------------------------------------------------------------
extern "C" void kernel_launch(void* const* d_in, const int* in_sizes, int n_in,
                              void* d_out, int out_size, void* d_ws, size_t ws_size,
                              hipStream_t stream) {
  const float* q   = (const float*)d_in[0];
  const float* k   = (const float*)d_in[1];
  const float* v   = (const float*)d_in[2];
  // d_in[3] = mask (present -> reference applies no masking; ignored)
  const float* wre = (const float*)d_in[4];
  const float* wim = (const float*)d_in[5];
  float* out = (float*)d_out;
  float* ws  = (float*)d_ws;   // needs 1 MB for twiddle tables

  twiddle_init<<<256, 256, 0, stream>>>(ws);
  spectral_kernel<<<B_ * H_, 256, 0, stream>>>(q, wre, wim, ws, out);
  attn_kernel<<<B_ * H_ * 8, 256, 0, stream>>>(q, k, v, out);
}